// FeatureShader_38783554682892
// MI455X (gfx1250) — compile-verified
//
#include <hip/hip_runtime.h>
#include <hip/hip_bf16.h>

// SoftRasterizer softmax-RGB blend, MI455X (gfx1250, wave32).
// HBM-bound streamer (~437 MB traffic -> ~19 us floor at 23.3 TB/s).
// CDNA5 path: the bulky colors record (96 B/pixel, consumed last) is staged
// via the async global->LDS data mover (ASYNCcnt), overlapping with the
// sigmoid/max VALU chain that depends only on the small direct-loaded records.

namespace {
constexpr float kSigma    = 1e-4f;
constexpr float kGamma    = 1e-4f;
constexpr float kZFar     = 100.0f;
constexpr float kZNear    = 1.0f;
constexpr float kEps      = 1e-10f;
constexpr int   kK        = 8;
constexpr float kInvSigma = 1.0f / kSigma;               // 1e4
constexpr float kInvGamma = 1.0f / kGamma;               // 1e4
constexpr float kInvZRng  = 1.0f / (kZFar - kZNear);     // 1/99
constexpr int   kThreads  = 256;                         // 8 wave32 waves
// LDS stride per lane: 96 B of colors padded to 112 B (28 dwords) so that
// ds_load_b128 reads are only 2-way bank conflicted (gcd(28,64)=4).
constexpr int   kLdsStrideF = 28;                        // floats per lane
}  // namespace

__global__ __launch_bounds__(kThreads) void softras_blend_kernel(
    const float4* __restrict__ colors4,   // (P, 6)  24 floats / pixel
    const float4* __restrict__ dists4,    // (P, 2)   8 floats / pixel
    const float4* __restrict__ zbuf4,     // (P, 2)   8 floats / pixel
    const int4*   __restrict__ p2f4,      // (P, 2)   8 int32  / pixel
    const float*  __restrict__ background,// (3,)
    float4*       __restrict__ out4,      // (P,)    rgba
    int npix)
{
    __shared__ __align__(16) float smem[kThreads * kLdsStrideF];  // 28 KB

    const int pix = blockIdx.x * blockDim.x + threadIdx.x;
    if (pix >= npix) return;
    const size_t p = (size_t)pix;

    // ---- stage colors into LDS via the CDNA5 async data mover ------------
    // Per ISA 08 §4.4 (async global->LDS load addressing),
    // i.e. the instruction offset advances BOTH addresses together.
    const unsigned int lds_addr =
        (unsigned int)(uintptr_t)(&smem[0]) +
        (unsigned int)threadIdx.x * (unsigned int)(kLdsStrideF * 4);
    const unsigned long long gaddr =
        (unsigned long long)(uintptr_t)(const void*)(colors4 + p * 6);

    asm volatile("global_load_async_to_lds_b128 %0, %1, off"
                 :: "v"(lds_addr), "v"(gaddr) : "memory");
    asm volatile("global_load_async_to_lds_b128 %0, %1, off offset:16"
                 :: "v"(lds_addr), "v"(gaddr) : "memory");
    asm volatile("global_load_async_to_lds_b128 %0, %1, off offset:32"
                 :: "v"(lds_addr), "v"(gaddr) : "memory");
    asm volatile("global_load_async_to_lds_b128 %0, %1, off offset:48"
                 :: "v"(lds_addr), "v"(gaddr) : "memory");
    asm volatile("global_load_async_to_lds_b128 %0, %1, off offset:64"
                 :: "v"(lds_addr), "v"(gaddr) : "memory");
    asm volatile("global_load_async_to_lds_b128 %0, %1, off offset:80"
                 :: "v"(lds_addr), "v"(gaddr) : "memory");

    // ---- latency-critical small records: direct global->VGPR b128 loads --
    const float4 d0 = dists4[p * 2 + 0];
    const float4 d1 = dists4[p * 2 + 1];
    const float4 z0 = zbuf4[p * 2 + 0];
    const float4 z1 = zbuf4[p * 2 + 1];
    const int4   f0 = p2f4[p * 2 + 0];
    const int4   f1 = p2f4[p * 2 + 1];

    const float d[kK] = {d0.x, d0.y, d0.z, d0.w, d1.x, d1.y, d1.z, d1.w};
    const float z[kK] = {z0.x, z0.y, z0.z, z0.w, z1.x, z1.y, z1.z, z1.w};
    const int   f[kK] = {f0.x, f0.y, f0.z, f0.w, f1.x, f1.y, f1.z, f1.w};

    // ---- coverage prob, alpha product, normalized inverse depth, max -----
    // (overlaps with the colors DMA still in flight)
    float prob[kK];
    float zi[kK];
    float alpha = 1.0f;        // prod(1 - prob)
    float zmax  = kEps;        // clip(max(z_inv), EPS) == max(EPS, max(z_inv))
#pragma unroll
    for (int k = 0; k < kK; ++k) {
        const bool m = (f[k] >= 0);
        // sigmoid(-d/sigma) = 1 / (1 + exp(d/sigma)); d/sigma in [0, 10]
        const float pr = m ? (1.0f / (1.0f + __expf(d[k] * kInvSigma))) : 0.0f;
        prob[k] = pr;
        alpha  *= (1.0f - pr);
        const float zv = m ? ((kZFar - z[k]) * kInvZRng) : 0.0f;
        zi[k] = zv;
        zmax  = fmaxf(zmax, zv);
    }

    const float delta = fmaxf(__expf((kEps - zmax) * kInvGamma), kEps);

    // ---- colors DMA must be resident before the blend loop ---------------
    asm volatile("s_wait_asynccnt 0x0" ::: "memory");
    const float4* cl =
        (const float4*)(&smem[threadIdx.x * kLdsStrideF]);
    float4 c[6];
#pragma unroll
    for (int i = 0; i < 6; ++i) c[i] = cl[i];   // ds_load_b128 x6
    const float* cc = reinterpret_cast<const float*>(c);

    // ---- softmax-in-depth weights + weighted color accumulation ----------
    float denom = delta;
    float wr = 0.0f, wg = 0.0f, wb = 0.0f;
#pragma unroll
    for (int k = 0; k < kK; ++k) {
        const float w = prob[k] * __expf((zi[k] - zmax) * kInvGamma);
        denom += w;
        wr = fmaf(w, cc[3 * k + 0], wr);
        wg = fmaf(w, cc[3 * k + 1], wg);
        wb = fmaf(w, cc[3 * k + 2], wb);
    }

    const float bg0 = background[0];
    const float bg1 = background[1];
    const float bg2 = background[2];
    const float inv = 1.0f / denom;

    float4 o;
    o.x = (wr + delta * bg0) * inv;
    o.y = (wg + delta * bg1) * inv;
    o.z = (wb + delta * bg2) * inv;
    o.w = 1.0f - alpha;
    out4[p] = o;   // global_store_b128
}

extern "C" void kernel_launch(void* const* d_in, const int* in_sizes, int n_in,
                              void* d_out, int out_size, void* d_ws, size_t ws_size,
                              hipStream_t stream) {
    (void)n_in; (void)out_size; (void)d_ws; (void)ws_size;

    const float* colors = (const float*)d_in[0];   // (N,H,W,K,3) f32
    const float* dists  = (const float*)d_in[1];   // (N,H,W,K)   f32
    const float* zbuf   = (const float*)d_in[2];   // (N,H,W,K)   f32
    const int*   p2f    = (const int*)d_in[3];     // (N,H,W,K)   int32
    const float* bg     = (const float*)d_in[4];   // (3,)        f32

    const int npix   = in_sizes[1] / kK;           // N*H*W from dists
    const int blocks = (npix + kThreads - 1) / kThreads;

    softras_blend_kernel<<<blocks, kThreads, 0, stream>>>(
        (const float4*)colors, (const float4*)dists, (const float4*)zbuf,
        (const int4*)p2f, bg, (float4*)d_out, npix);
}